// DifferentiableAstar_41747082117708
// MI455X (gfx1250) — compile-verified
//
#include <hip/hip_runtime.h>
#include <math.h>

#ifndef __has_builtin
#define __has_builtin(x) 0
#endif

#if __has_builtin(__builtin_amdgcn_tensor_load_to_lds)
#define HAVE_TDM 1
#else
#define HAVE_TDM 0
#endif

#if __has_builtin(__builtin_amdgcn_wmma_f32_16x16x4_f32)
#define HAVE_WMMA 1
#else
#define HAVE_WMMA 0
#endif

#if defined(__has_include)
#if __has_include(<hip/amd_detail/amd_gfx1250_TDM.h>)
#define TDM_6ARG 1
#else
#define TDM_6ARG 0
#endif
#else
#define TDM_6ARG 0
#endif

#define BB  16      // batch
#define NC  1024    // cells per batch map (32x32)
#define TPB 1024    // one workgroup: 32 waves, 2 waves (64 threads) per batch
#define CPT 16      // cells per thread (stride 64)

typedef unsigned int u32;
typedef u32 u32x4 __attribute__((ext_vector_type(4)));
typedef int i32x4 __attribute__((ext_vector_type(4)));
typedef int i32x8 __attribute__((ext_vector_type(8)));
typedef float v2f __attribute__((ext_vector_type(2)));
typedef float v8f __attribute__((ext_vector_type(8)));

__device__ __forceinline__ float clip01(float x) {
    // NaN-propagating clip, matching jnp.clip semantics
    x = x < 0.0f ? 0.0f : x;
    x = x > 1.0f ? 1.0f : x;
    return x;
}

// exp(-(G*g + (1-G)*h)/sqrt(W)) with the reference's expression order
__device__ __forceinline__ float expterm(float gv, float hv) {
#pragma clang fp contract(off)
    const float f = 0.5f * gv + 0.5f * hv;
    return expf((-f) * 0.17677669529663689f);
}

__launch_bounds__(TPB, 1)
__global__ void astar1250_kernel(const float* __restrict__ cm,
                                 const float* __restrict__ sm,
                                 const float* __restrict__ gm,
                                 const float* __restrict__ om,
                                 float* __restrict__ out)
{
#pragma clang fp contract(off)
    __shared__ float lds_cm [BB * NC];   // 64 KB cost maps (TDM-staged)
    __shared__ float lds_g  [BB * NC];   // 64 KB g values
    __shared__ float lds_par[BB * NC];   // 64 KB parents (float, blended!)
    __shared__ float lds_pv[32];
    __shared__ int   lds_pi[32];
    __shared__ float lds_ps[32];
    __shared__ int   lds_goal[BB];
    __shared__ int   lds_sel[BB];
#if HAVE_WMMA
    __shared__ float lds_glr[BB];        // einsum goal_loc row coords
    __shared__ float lds_glc[BB];        // einsum goal_loc col coords
#endif

    const int tid  = (int)threadIdx.x;
    const int b    = tid >> 6;       // batch group (16 groups x 64 threads)
    const int j    = tid & 63;       // lane within group
    const int wave = tid >> 5;       // 0..31
    const int lane = tid & 31;
    const int base = b * NC;

#if HAVE_TDM
    if (wave == 0) {
        // --- Tensor Data Mover: stage all 16 cost maps (1024 x 16 f32) to LDS
        u32 ldsa = (u32)(size_t)(&lds_cm[0]);
        unsigned long long ga = (unsigned long long)(size_t)cm;
        u32x4 g0 = { 1u,                                     // count=1, user desc
                     ldsa,                                   // lds_addr
                     (u32)ga,                                // global addr lo
                     (u32)((ga >> 32) & 0x1FFFFFFull) | 0x80000000u }; // hi + type=2
        i32x8 g1 = { (int)0x00020000u,   // workgroup_mask=0, data_size=4B
                     (int)0x04000000u,   // tensor_dim0 = 1024
                     (int)0x00100000u,   // tensor_dim1 = 16
                     (int)0x04000000u,   // tile_dim0   = 1024
                     16,                 // tile_dim1 = 16, tile_dim2 = 0
                     1024,               // tensor_dim0_stride = 1024 (lo32)
                     (int)0x40000000u,   // dim1_stride lo16 = 16384
                     0 };
        i32x4 g2 = { 0, 0, 0, 0 };
        i32x4 g3 = { 0, 0, 0, 0 };
#if TDM_6ARG
        i32x8 g4 = { 0, 0, 0, 0, 0, 0, 0, 0 };
        __builtin_amdgcn_tensor_load_to_lds(g0, g1, g2, g3, g4, 0);
#else
        __builtin_amdgcn_tensor_load_to_lds(g0, g1, g2, g3, 0);
#endif
    }
#endif

    __builtin_prefetch(gm + base + j, 0, 2);
    __builtin_prefetch(om + base + j, 0, 2);

    // ---- per-cell persistent register state
    u32 ommask = 0;
    float open_[CPT];
    float hist_[CPT];
    float hmap_[CPT];
    float e_[CPT];        // cached exp term (depends only on g and h)

#pragma unroll
    for (int k = 0; k < CPT; ++k) {
        const int c = j + (k << 6);
        const float gmv = gm[base + c];
        const float smv = sm[base + c];
        const float omv = om[base + c];
        if (omv > 0.5f) ommask |= (1u << k);
        open_[k] = smv;          // open init = start map (exact floats)
        hist_[k] = 0.0f;
        lds_g[base + c] = 0.0f;
        if (gmv > 0.5f) lds_goal[b] = c;   // one-hot: single writer per batch
    }
#if HAVE_WMMA
    if (tid < BB) { lds_glr[tid] = 0.0f; lds_glc[tid] = 0.0f; }

    // ---- goal_loc = einsum('kij,bij->bk', loc, gm) as a real matrix product:
    // (16 batches) x (2 coords) x (K=1024 cells), V_WMMA_F32_16X16X4_F32.
    // Exact in any summation order (one-hot times small ints), so per-wave
    // K-partials combined with ds_add_f32 are bit-identical to the einsum.
    v8f acc = { 0.0f, 0.0f, 0.0f, 0.0f, 0.0f, 0.0f, 0.0f, 0.0f };
    {
        const int m  = lane & 15;          // A row (batch) / B col (coord)
        const int kk = (lane >> 4) * 2;    // per-lane K split
#pragma unroll
        for (int ch = 0; ch < 8; ++ch) {
            const int c0 = wave * 32 + ch * 4;
            const int k0 = c0 + kk, k1 = k0 + 1;
            v2f a, bm;
            a.x = gm[m * NC + k0];
            a.y = gm[m * NC + k1];
            bm.x = (m == 0) ? (float)(k0 >> 5) : (m == 1) ? (float)(k0 & 31) : 0.0f;
            bm.y = (m == 0) ? (float)(k1 >> 5) : (m == 1) ? (float)(k1 & 31) : 0.0f;
            acc = __builtin_amdgcn_wmma_f32_16x16x4_f32(false, a, false, bm,
                                                        (short)0, acc, false, false);
        }
    }
#endif

#if HAVE_TDM
    if (wave == 0) {
#if __has_builtin(__builtin_amdgcn_s_wait_tensorcnt)
        __builtin_amdgcn_s_wait_tensorcnt(0);
#else
        asm volatile("s_wait_tensorcnt 0" ::: "memory");
#endif
    }
#else
    for (int idx = tid; idx < BB * NC; idx += TPB) lds_cm[idx] = cm[idx];
#endif
    __syncthreads();   // lds_cm, lds_goal, lds_g, glr/glc zero-init visible

#if HAVE_WMMA
    // D[b][n]: n in lanes (b<8 -> lane n, b>=8 -> lane 16+n), VGPR r = b&7
    if (lane == 0 || lane == 1 || lane == 16 || lane == 17) {
        const int mb = (lane >> 4) * 8;
        float* dst = ((lane & 15) == 0) ? lds_glr : lds_glc;
#pragma unroll
        for (int r = 0; r < 8; ++r) atomicAdd(&dst[mb + r], acc[r]);
    }
    __syncthreads();
#endif

    const int goalIdx = lds_goal[b];
    {
#if HAVE_WMMA
        const float gi = lds_glr[b];       // == (float)(goalIdx>>5) exactly
        const float gj = lds_glc[b];
#else
        const float gi = (float)(goalIdx >> 5);
        const float gj = (float)(goalIdx & 31);
#endif
#pragma unroll
        for (int k = 0; k < CPT; ++k) {
            const int c = j + (k << 6);
            const float ci = (float)(c >> 5), cj = (float)(c & 31);
            const float dx = fabsf(ci - gi), dy = fabsf(cj - gj);
            const float mn = dx < dy ? dx : dy;
            const float hh = (dx + dy) - mn;
            const float euc = sqrtf(dx * dx + dy * dy);
            hmap_[k] = (hh + 0.001f * euc) + lds_cm[base + c];   // h = heuristic + cm
            e_[k] = expterm(0.0f, hmap_[k]);                     // g = 0 initially
            lds_par[base + c] = (float)goalIdx;                  // parents0
        }
    }
    __syncthreads();

    int tstep = NC - 1;

    for (int it = 0; it < NC; ++it) {
        // ---- local argmax (first-index tie-break) + local sum of f_exp
        float bv = -1.0f; int bi = 0; float lsum = 0.0f;
#pragma unroll
        for (int k = 0; k < CPT; ++k) {
            const float fe = e_[k] * open_[k];
            if (fe > bv) { bv = fe; bi = j + (k << 6); }
            lsum += fe;
        }
        // ---- wave32 butterfly: argmax with min-index tie-break, plus sum
#pragma unroll
        for (int off = 16; off >= 1; off >>= 1) {
            const float ov = __shfl_xor(bv, off, 32);
            const int   oi = __shfl_xor(bi, off, 32);
            const float os = __shfl_xor(lsum, off, 32);
            lsum += os;
            if (ov > bv || (ov == bv && oi < bi)) { bv = ov; bi = oi; }
        }
        if (lane == 0) { lds_pv[wave] = bv; lds_pi[wave] = bi; lds_ps[wave] = lsum; }
        __syncthreads();                                   // B1

        // ---- combine the group's two waves
        const int w0 = b * 2, w1 = b * 2 + 1;
        const float v0 = lds_pv[w0], v1 = lds_pv[w1];
        const int   i0 = lds_pi[w0], i1 = lds_pi[w1];
        const float ssum = lds_ps[w0] + lds_ps[w1];
        int s; float vs;
        if (v1 > v0 || (v1 == v0 && i1 < i0)) { s = i1; vs = v1; }
        else                                  { s = i0; vs = v0; }

        // straight-through softmax forward value at the argmax:
        // snm = (1 - y_s) + y_s  (usually exactly 1.0, sometimes 1-2^-24)
        const float ys  = vs / ssum;
        const float snm = (1.0f - ys) + ys;
        const float dist = (s == goalIdx) ? snm : 0.0f;
        const float unsolved = (dist < 1e-8f) ? 1.0f : 0.0f;
        if (j == 0) lds_sel[b] = (unsolved == 0.0f) ? 1 : 0;

        const float gs2 = (lds_g[base + s] + lds_cm[base + s]) * snm; // expand((g+cm)*snm)
        const int sr = s >> 5, sc = s & 31;
        const float sf = (float)s;

        // ---- per-cell updates (each cell owned by exactly one thread)
#pragma unroll
        for (int k = 0; k < CPT; ++k) {
            const int c = j + (k << 6);
            float op = open_[k];
            float hi = hist_[k];
            if (c == s) {
                hi = clip01(hi + snm);
                op = clip01(op - unsolved * snm);
            }
            const int cr = c >> 5, cc = c & 31;
            int dr = cr - sr; dr = dr < 0 ? -dr : dr;
            int dc = cc - sc; dc = dc < 0 ? -dc : dc;
            const bool nbr = (dr <= 1) && (dc <= 1) && (c != s);
            if (nbr) {
                const float omv = ((ommask >> k) & 1u) ? 1.0f : 0.0f;
                const float nb  = snm * omv;                       // expand(snm)*om
                const float gv  = lds_g[base + c];
                const float gt  = (gv > gs2) ? 1.0f : 0.0f;
                const float idxv = ((1.0f - op) * (1.0f - hi) + op * gt) * nb;
                const float gn  = gs2 * idxv + gv * (1.0f - idxv);
                lds_g[base + c]   = gn;
                e_[k] = expterm(gn, hmap_[k]);                     // refresh cached exp
                op = clip01(op + idxv);
                lds_par[base + c] = sf * idxv + lds_par[base + c] * (1.0f - idxv);
            }
            open_[k] = op;
            hist_[k] = hi;
        }

        __syncthreads();                                   // B2: lds_sel + LDS updates
        int all = 1;
#pragma unroll
        for (int q = 0; q < BB; ++q) all &= lds_sel[q];
        if (all) { tstep = it; break; }                    // uniform across block
    }

    // ---- outputs: histories [0,16384), path [16384,32768)
#pragma unroll
    for (int k = 0; k < CPT; ++k) {
        const int c = j + (k << 6);
        out[base + c] = hist_[k];
        out[BB * NC + base + c] = 0.0f;
    }
    __syncthreads();

    if (tid < BB) {
        const int bb = tid;
        const int gI = lds_goal[bb];
        float* path = out + BB * NC + bb * NC;
        path[gI] = 1.0f;
        int loc = (int)lds_par[bb * NC + gI];          // trunc == astype(int32)
        loc = loc < 0 ? 0 : (loc > NC - 1 ? NC - 1 : loc);
        for (int q = 0; q < tstep; ++q) {
            path[loc] = 1.0f;
            int nxt = (int)lds_par[bb * NC + loc];
            loc = nxt < 0 ? 0 : (nxt > NC - 1 ? NC - 1 : nxt);  // XLA-style clamp
        }
    }
}

extern "C" void kernel_launch(void* const* d_in, const int* in_sizes, int n_in,
                              void* d_out, int out_size, void* d_ws, size_t ws_size,
                              hipStream_t stream) {
    (void)in_sizes; (void)n_in; (void)out_size; (void)d_ws; (void)ws_size;
    const float* cm = (const float*)d_in[0];
    const float* sm = (const float*)d_in[1];
    const float* gm = (const float*)d_in[2];
    const float* om = (const float*)d_in[3];
    float* out = (float*)d_out;
    hipLaunchKernelGGL(astar1250_kernel, dim3(1), dim3(TPB), 0, stream,
                       cm, sm, gm, om, out);
}